// PME_24249385353282
// MI455X (gfx1250) — compile-verified
//
#include <hip/hip_runtime.h>
#include <math.h>

// ---------------------------------------------------------------------------
// Ewald reciprocal-space energy, factorized as a real GEMM for CDNA5 WMMA.
//   E = (2*pi/V) * sum_{k!=0} exp(-|k|^2/(4a^2))/|k|^2 * |S(k)|^2
//   S(h,k,l) = sum_i q_i e^{i(h u_i + k v_i + l w_i)},  (u,v,w) = 2pi*frac coords
// S is computed as M1(64x4096) @ B(4096x1280) with fp32 WMMA 16x16x4.
// B is stored K-contiguous (M2T[col][k]) so each lane's (K,K+1) operand pair
// is a single global_load_b64 -> 2 loads + 1 wmma per K=4 step.
// ---------------------------------------------------------------------------

#define NPART   4096
#define KMAX    12
#define NH      25            // 2*KMAX+1
#define NKL     625           // NH*NH  (k,l) pairs
#define M1_ROWS 64            // pad of 2*NH = 50
#define M2_COLS 1280          // pad of 2*NKL = 1250
#define N_TILES (M2_COLS/16)  // 80
#define M_TILES (M1_ROWS/16)  // 4
#define EWALD_ALPHA 0.34f
#define TWO_PI_F 6.2831853071795864769f

// workspace layout in floats
#define OFF_M1 0
#define OFF_VW (OFF_M1 + M1_ROWS * NPART)                   // per-particle (v,w)
#define OFF_M2 (OFF_VW + 2 * NPART)                         // M2T[col][k], col-major cols
#define OFF_P  (OFF_M2 + (size_t)NPART * M2_COLS)
#define WS_FLOATS (OFF_P + (size_t)M1_ROWS * M2_COLS)
#define WS_BYTES  (WS_FLOATS * sizeof(float))               // ~22.4 MB

typedef float v2f __attribute__((ext_vector_type(2)));
typedef float v8f __attribute__((ext_vector_type(8)));

__device__ __forceinline__ void inv3x3(const float* __restrict__ B, float inv[9]) {
    float a = B[0], b = B[1], c = B[2];
    float d = B[3], e = B[4], f = B[5];
    float g = B[6], h = B[7], i = B[8];
    float A  =  (e * i - f * h);
    float Bc = -(d * i - f * g);
    float C  =  (d * h - e * g);
    float det = a * A + b * Bc + c * C;
    float id = 1.0f / det;
    inv[0] = A * id;             inv[1] = -(b * i - c * h) * id;  inv[2] =  (b * f - c * e) * id;
    inv[3] = Bc * id;            inv[4] =  (a * i - c * g) * id;  inv[5] = -(a * f - c * d) * id;
    inv[6] = C * id;             inv[7] = -(a * h - b * g) * id;  inv[8] =  (a * e - b * d) * id;
}

// ---------------------------------------------------------------------------
// Kernel 1: per-particle h-tables + (v,w) phases.
//   M1[h'][i] = q_i*cos((h'-K)u_i) (rows 0..24), M1[25+h'][i] = q_i*sin(..),
//   rows 50..63 zero padding.  VW[i] = (v_i, w_i).  All writes coalesced in i.
// ---------------------------------------------------------------------------
__global__ void k_build_m1(const float* __restrict__ coords,
                           const float* __restrict__ box,
                           const float* __restrict__ q,
                           float* __restrict__ ws) {
    int i = blockIdx.x * blockDim.x + threadIdx.x;
    if (i >= NPART) return;
    float Bi[9];
    inv3x3(box, Bi);
    float x = coords[3 * i], y = coords[3 * i + 1], z = coords[3 * i + 2];
    // (u,v,w) = 2pi * (columns of inv(box)) . r
    float u = TWO_PI_F * (Bi[0] * x + Bi[3] * y + Bi[6] * z);
    float v = TWO_PI_F * (Bi[1] * x + Bi[4] * y + Bi[7] * z);
    float w = TWO_PI_F * (Bi[2] * x + Bi[5] * y + Bi[8] * z);
    ws[OFF_VW + 2 * i]     = v;
    ws[OFF_VW + 2 * i + 1] = w;

    float qi = q[i];
    float su, cu, sr, cr;
    sincosf(u, &su, &cu);
    sincosf(-(float)KMAX * u, &sr, &cr);   // start at h = -K, rotate by e^{iu}
    float* __restrict__ M1 = ws + OFF_M1;
#pragma unroll
    for (int hp = 0; hp < NH; ++hp) {
        M1[hp * NPART + i]        = qi * cr;
        M1[(NH + hp) * NPART + i] = qi * sr;
        float c2 = cr * cu - sr * su;
        sr = cr * su + sr * cu;
        cr = c2;
    }
#pragma unroll
    for (int r = 2 * NH; r < M1_ROWS; ++r) M1[r * NPART + i] = 0.0f;
}

// ---------------------------------------------------------------------------
// Kernel 2: one wave per (k,l) pair (640 incl. padding).  Lanes stride over
// particles: phi = k*v_i + l*w_i; store cos/sin K-contiguous:
//   M2T[(2j)*NPART + i], M2T[(2j+1)*NPART + i]   (coalesced across lanes).
// Pairs 625..639 fill the padded columns with zeros.
// ---------------------------------------------------------------------------
__global__ void k_build_m2(float* __restrict__ ws) {
    int wave = (int)((blockIdx.x * blockDim.x + threadIdx.x) >> 5);  // pair j
    int lane = threadIdx.x & 31;
    const float* __restrict__ VW = ws + OFF_VW;
    float* __restrict__ colC = ws + OFF_M2 + (size_t)(2 * wave) * NPART;
    float* __restrict__ colS = colC + NPART;
    if (wave < NKL) {
        int kp = wave / NH, lp = wave - kp * NH;
        float fk = (float)(kp - KMAX), fl = (float)(lp - KMAX);
        for (int i = lane; i < NPART; i += 32) {
            float v = VW[2 * i], w = VW[2 * i + 1];
            float s, c;
            sincosf(fk * v + fl * w, &s, &c);
            colC[i] = c;
            colS[i] = s;
        }
    } else {
        for (int i = lane; i < NPART; i += 32) {
            colC[i] = 0.0f;
            colS[i] = 0.0f;
        }
    }
}

// ---------------------------------------------------------------------------
// Kernel 3: P(64x1280) = M1(64x4096) @ B(4096x1280) via V_WMMA_F32_16X16X4_F32.
// Exactly 320 waves (40 blocks x 8 waves), one 16x16 tile per wave, EXEC all 1s.
// A/B lane layout per ISA 7.12.2: VGPR = K&1, lane-group = K>>1, in-group = M/N.
// Both operands are single b64 loads (K-pair contiguous).
// ---------------------------------------------------------------------------
__global__ void __launch_bounds__(256) k_gemm(float* __restrict__ ws) {
    const float* __restrict__ M1  = ws + OFF_M1;
    const float* __restrict__ M2T = ws + OFF_M2;
    float* __restrict__ P = ws + OFF_P;

    int wave = (int)((blockIdx.x * blockDim.x + threadIdx.x) >> 5);  // 0..319
    int lane = threadIdx.x & 31;
    int mt = wave / N_TILES;            // 0..3
    int nt = wave - mt * N_TILES;       // 0..79
    int g = lane >> 4;                  // lane group (K half)
    int r = lane & 15;                  // row (A) / col (B) within tile

    const float* __restrict__ arow = M1  + (size_t)(mt * 16 + r) * NPART + 2 * g;
    const float* __restrict__ brow = M2T + (size_t)(nt * 16 + r) * NPART + 2 * g;

    v8f acc = {};
#pragma unroll 8
    for (int kb = 0; kb < NPART; kb += 4) {
        v2f a = *(const v2f*)(arow + kb);   // A[m][kb+2g], A[m][kb+2g+1]
        v2f b = *(const v2f*)(brow + kb);   // B[kb+2g][n], B[kb+2g+1][n]
        acc = __builtin_amdgcn_wmma_f32_16x16x4_f32(
            /*neg_a=*/false, a, /*neg_b=*/false, b,
            /*c_mod=*/(short)0, acc, /*reuse_a=*/false, /*reuse_b=*/false);
    }
#pragma unroll
    for (int vv = 0; vv < 8; ++vv)
        P[(size_t)(mt * 16 + vv + 8 * g) * M2_COLS + nt * 16 + r] = acc[vv];
}

// ---------------------------------------------------------------------------
// Kernel 4: E = (2pi/V) * sum_{k!=0} fac(k) * (Sre^2 + Sim^2)
//   Sre = P[h, 2j] - P[25+h, 2j+1] ;  Sim = P[h, 2j+1] + P[25+h, 2j]
// ---------------------------------------------------------------------------
__global__ void k_reduce(const float* __restrict__ box,
                         const float* __restrict__ ws,
                         float* __restrict__ out) {
    const float* __restrict__ P = ws + OFF_P;
    float Bi[9];
    inv3x3(box, Bi);
    // reciprocal rows: R[m][j] = 2pi * Binv[j][m]
    float R00 = TWO_PI_F * Bi[0], R01 = TWO_PI_F * Bi[3], R02 = TWO_PI_F * Bi[6];
    float R10 = TWO_PI_F * Bi[1], R11 = TWO_PI_F * Bi[4], R12 = TWO_PI_F * Bi[7];
    float R20 = TWO_PI_F * Bi[2], R21 = TWO_PI_F * Bi[5], R22 = TWO_PI_F * Bi[8];
    float a = box[0], b = box[1], c = box[2];
    float d = box[3], e = box[4], f = box[5];
    float gg = box[6], h = box[7], ii = box[8];
    float det = a * (e * ii - f * h) - b * (d * ii - f * gg) + c * (d * h - e * gg);
    float vol = fabsf(det);
    const float inv4a2 = 1.0f / (4.0f * EWALD_ALPHA * EWALD_ALPHA);

    float local = 0.0f;
    const int total = NH * NH * NH;  // 15625
    for (int idx = blockIdx.x * blockDim.x + threadIdx.x; idx < total;
         idx += gridDim.x * blockDim.x) {
        int hp  = idx / (NH * NH);
        int rem = idx - hp * (NH * NH);
        int kp  = rem / NH;
        int lp  = rem - kp * NH;
        float fh = (float)(hp - KMAX), fk = (float)(kp - KMAX), fl = (float)(lp - KMAX);
        float kx = fh * R00 + fk * R10 + fl * R20;
        float ky = fh * R01 + fk * R11 + fl * R21;
        float kz = fh * R02 + fk * R12 + fl * R22;
        float k2 = kx * kx + ky * ky + kz * kz;
        if (k2 > 0.0f) {
            float Pa = P[(size_t)hp * M2_COLS + 2 * rem];
            float Pb = P[(size_t)hp * M2_COLS + 2 * rem + 1];
            float Pc = P[(size_t)(NH + hp) * M2_COLS + 2 * rem];
            float Pd = P[(size_t)(NH + hp) * M2_COLS + 2 * rem + 1];
            float Sre = Pa - Pd;
            float Sim = Pb + Pc;
            float fac = expf(-k2 * inv4a2) / k2;
            local += fac * (Sre * Sre + Sim * Sim);
        }
    }

    __shared__ float red[256];
    red[threadIdx.x] = local;
    __syncthreads();
    for (int s = 128; s > 0; s >>= 1) {
        if (threadIdx.x < s) red[threadIdx.x] += red[threadIdx.x + s];
        __syncthreads();
    }
    if (threadIdx.x == 0) atomicAdd(out, red[0] * (TWO_PI_F / vol));
}

// ---------------------------------------------------------------------------
extern "C" void kernel_launch(void* const* d_in, const int* in_sizes, int n_in,
                              void* d_out, int out_size, void* d_ws, size_t ws_size,
                              hipStream_t stream) {
    (void)in_sizes; (void)n_in; (void)out_size;
    const float* coords = (const float*)d_in[0];  // (4096,3) f32
    const float* box    = (const float*)d_in[1];  // (3,3)    f32
    const float* q      = (const float*)d_in[2];  // (4096,)  f32
    float* out = (float*)d_out;
    float* ws  = (float*)d_ws;

    hipMemsetAsync(d_out, 0, sizeof(float), stream);
    if (ws_size < WS_BYTES) return;  // workspace too small; deterministic no-op (out=0)

    k_build_m1<<<NPART / 256, 256, 0, stream>>>(coords, box, q, ws);
    k_build_m2<<<(M2_COLS / 2) / 8, 256, 0, stream>>>(ws);           // 640 waves exactly
    k_gemm<<<(M_TILES * N_TILES) / 8, 256, 0, stream>>>(ws);         // 320 waves exactly
    k_reduce<<<64, 256, 0, stream>>>(box, ws, out);
}